// InstaSHAP_74715251081636
// MI455X (gfx1250) — compile-verified
//
#include <hip/hip_runtime.h>

typedef float v2f __attribute__((ext_vector_type(2)));
typedef float v8f __attribute__((ext_vector_type(8)));

#define B_SZ 8192
#define F_SZ 256
#define M_SZ 512
#define K_SZ 8
#define H_SZ 16

#define ROWS_PER_BLOCK 32   // one wave, two 16-row WMMA tiles
#define THREADS 32
#define MSPLIT 4
#define M_PER (M_SZ / MSPLIT)

__global__ void zero_out_kernel(float* __restrict__ out, int n) {
    int i = blockIdx.x * blockDim.x + threadIdx.x;
    if (i < n) out[i] = 0.0f;
}

__global__ __launch_bounds__(THREADS)
void instashap_wmma_kernel(const float* __restrict__ x,
                           const float* __restrict__ S,
                           const int*   __restrict__ idx,   // (M,K) int32
                           const float* __restrict__ W1,    // (M,K,H)
                           const float* __restrict__ b1,    // (M,H)
                           const float* __restrict__ W2,    // (M,H)
                           const float* __restrict__ b2,    // (M,)
                           float* __restrict__ out)         // (B,)
{
    __shared__ float         xs[ROWS_PER_BLOCK * F_SZ];   // 32 KB
    __shared__ unsigned char ssh[ROWS_PER_BLOCK * F_SZ];  // 8 KB

    const int lane = threadIdx.x & 31;
    const int b0   = blockIdx.x * ROWS_PER_BLOCK;

    // ---- Stage x rows (f32) and S rows (0/1 bytes) into LDS ----
    {
        const float4* xv = (const float4*)(x + (size_t)b0 * F_SZ);
        const float4* sv = (const float4*)(S + (size_t)b0 * F_SZ);
        float4*       xd = (float4*)xs;
        uchar4*       sd = (uchar4*)ssh;
        #pragma unroll 4
        for (int i = lane; i < ROWS_PER_BLOCK * F_SZ / 4; i += THREADS) {
            xd[i] = xv[i];
            float4 s = sv[i];
            sd[i] = make_uchar4((unsigned char)s.x, (unsigned char)s.y,
                                (unsigned char)s.z, (unsigned char)s.w);
        }
    }
    __syncthreads();

    const int lo = lane & 15;     // A row (within tile) / D column
    const int hi = lane >> 4;     // lane-half: K phase / D row half

    const float*         xrow0 = xs  + lo * F_SZ;          // tile 0 rows 0..15
    const float*         xrow1 = xs  + (16 + lo) * F_SZ;   // tile 1 rows 16..31
    const unsigned char* srow0 = ssh + lo * F_SZ;
    const unsigned char* srow1 = ssh + (16 + lo) * F_SZ;

    v8f   Acc0 = {0.f,0.f,0.f,0.f,0.f,0.f,0.f,0.f};
    v8f   Acc1 = {0.f,0.f,0.f,0.f,0.f,0.f,0.f,0.f};
    float accB2_0 = 0.f, accB2_1 = 0.f;

    const int   mBase = blockIdx.y * M_PER;
    const int4* idxq  = (const int4*)idx;

    for (int mi = 0; mi < M_PER; ++mi) {
        const int m = mBase + mi;

        int4 q0 = idxq[2 * m + 0];               // idx[m][0..3] (uniform)
        int4 q1 = idxq[2 * m + 1];               // idx[m][4..7]

        int mp = (m + 8 < M_SZ) ? m + 8 : m;     // global_prefetch_b8
        __builtin_prefetch(W1 + (size_t)mp * (K_SZ * H_SZ), 0, 1);

        // gather columns for this lane: k = 4c + 2*hi + v
        int iA0 = hi ? q0.z : q0.x;
        int iA1 = hi ? q0.w : q0.y;
        int iA2 = hi ? q1.z : q1.x;
        int iA3 = hi ? q1.w : q1.y;

        // B operands: W1[m] 4x16 slices (v -> k = 4c + 2*hi + v, n = lo)
        const float* w1m = W1 + (size_t)m * (K_SZ * H_SZ);
        const int k0 = 2 * hi;
        v2f bm0 = { w1m[(k0 + 0) * H_SZ + lo], w1m[(k0 + 1) * H_SZ + lo] };
        v2f bm1 = { w1m[(4 + k0 + 0) * H_SZ + lo], w1m[(4 + k0 + 1) * H_SZ + lo] };

        float b1g = b1[m * H_SZ + lo];
        float w2g = W2[m * H_SZ + lo];
        float b2v = b2[m];

        // rank-1 bias chunk: D += act(r) * b1(n)   (A = act ones-col, B = b1 row)
        v2f bm2 = { hi ? 0.f : b1g, 0.f };

        int4 qs = hi ? q1 : q0;   // this half sums S over k = 4*hi .. 4*hi+3

        // ================= tile 0 (rows b0 .. b0+15) =================
        {
            int ssum = (int)srow0[qs.x] + (int)srow0[qs.y] +
                       (int)srow0[qs.z] + (int)srow0[qs.w];
            ssum += __shfl_xor(ssum, 16, 32);
            float p = (ssum == K_SZ) ? 1.0f : 0.0f;   // act for row lo

            v2f a0 = { xrow0[iA0] * p, xrow0[iA1] * p };   // act-scaled A rows
            v2f a1 = { xrow0[iA2] * p, xrow0[iA3] * p };
            v2f a2 = { hi ? 0.f : p, 0.f };

            v8f d = {0.f,0.f,0.f,0.f,0.f,0.f,0.f,0.f};
            d = __builtin_amdgcn_wmma_f32_16x16x4_f32(false, a2, false, bm2,
                                                      (short)0, d, false, false);
            d = __builtin_amdgcn_wmma_f32_16x16x4_f32(false, a0, false, bm0,
                                                      (short)0, d, false, false);
            d = __builtin_amdgcn_wmma_f32_16x16x4_f32(false, a1, false, bm1,
                                                      (short)0, d, false, false);
            #pragma unroll
            for (int j = 0; j < 8; ++j)
                Acc0[j] = fmaf(fmaxf(d[j], 0.0f), w2g, Acc0[j]);
            accB2_0 = fmaf(p, b2v, accB2_0);
        }
        // ================= tile 1 (rows b0+16 .. b0+31) ==============
        {
            int ssum = (int)srow1[qs.x] + (int)srow1[qs.y] +
                       (int)srow1[qs.z] + (int)srow1[qs.w];
            ssum += __shfl_xor(ssum, 16, 32);
            float p = (ssum == K_SZ) ? 1.0f : 0.0f;

            v2f a0 = { xrow1[iA0] * p, xrow1[iA1] * p };
            v2f a1 = { xrow1[iA2] * p, xrow1[iA3] * p };
            v2f a2 = { hi ? 0.f : p, 0.f };

            v8f d = {0.f,0.f,0.f,0.f,0.f,0.f,0.f,0.f};
            d = __builtin_amdgcn_wmma_f32_16x16x4_f32(false, a2, false, bm2,
                                                      (short)0, d, false, false);
            d = __builtin_amdgcn_wmma_f32_16x16x4_f32(false, a0, false, bm0,
                                                      (short)0, d, false, false);
            d = __builtin_amdgcn_wmma_f32_16x16x4_f32(false, a1, false, bm1,
                                                      (short)0, d, false, false);
            #pragma unroll
            for (int j = 0; j < 8; ++j)
                Acc1[j] = fmaf(fmaxf(d[j], 0.0f), w2g, Acc1[j]);
            accB2_1 = fmaf(p, b2v, accB2_1);
        }
    }

    // ---- once: reduce the 16 columns within each lane-half ----
    #pragma unroll
    for (int j = 0; j < 8; ++j) {
        float v0 = Acc0[j], v1 = Acc1[j];
        v0 += __shfl_xor(v0, 1, 32);  v1 += __shfl_xor(v1, 1, 32);
        v0 += __shfl_xor(v0, 2, 32);  v1 += __shfl_xor(v1, 2, 32);
        v0 += __shfl_xor(v0, 4, 32);  v1 += __shfl_xor(v1, 4, 32);
        v0 += __shfl_xor(v0, 8, 32);  v1 += __shfl_xor(v1, 8, 32);
        Acc0[j] = v0;  Acc1[j] = v1;
    }
    // bring hi-half row totals (rows j+8) to every lane
    float hv0[8], hv1[8];
    #pragma unroll
    for (int j = 0; j < 8; ++j) {
        hv0[j] = __shfl(Acc0[j], 16, 32);
        hv1[j] = __shfl(Acc1[j], 16, 32);
    }

    // writer lanes 0..15: lane L writes row L of each tile; accB2_* on lane L
    // is exactly act-weighted b2 for row L (predf was computed for row lo==L).
    if (lane < 16) {
        float v0 = 0.f, v1 = 0.f;
        #pragma unroll
        for (int j = 0; j < 8; ++j) {
            v0 = (lane == j)     ? Acc0[j] : v0;
            v0 = (lane == j + 8) ? hv0[j]  : v0;
            v1 = (lane == j)     ? Acc1[j] : v1;
            v1 = (lane == j + 8) ? hv1[j]  : v1;
        }
        atomicAdd(out + b0 + lane,      v0 + accB2_0);
        atomicAdd(out + b0 + 16 + lane, v1 + accB2_1);
    }
}

extern "C" void kernel_launch(void* const* d_in, const int* in_sizes, int n_in,
                              void* d_out, int out_size, void* d_ws, size_t ws_size,
                              hipStream_t stream) {
    (void)in_sizes; (void)n_in; (void)d_ws; (void)ws_size;
    const float* x   = (const float*)d_in[0];
    const float* S   = (const float*)d_in[1];
    // d_in[2] = transform_matrix (unused; idx carries the same information)
    const int*   idx = (const int*)d_in[3];
    const float* W1  = (const float*)d_in[4];
    const float* b1  = (const float*)d_in[5];
    const float* W2  = (const float*)d_in[6];
    const float* b2  = (const float*)d_in[7];
    float*       out = (float*)d_out;

    zero_out_kernel<<<(out_size + 255) / 256, 256, 0, stream>>>(out, out_size);

    dim3 grid(B_SZ / ROWS_PER_BLOCK, MSPLIT);
    instashap_wmma_kernel<<<grid, THREADS, 0, stream>>>(x, S, idx, W1, b1, W2, b2, out);
}